// RNN_18141941859084
// MI455X (gfx1250) — compile-verified
//
#include <hip/hip_runtime.h>

typedef __attribute__((ext_vector_type(2))) float v2f;
typedef __attribute__((ext_vector_type(4))) float v4f;
typedef __attribute__((ext_vector_type(8))) float v8f;

#define BATCH 64
#define SEQT  2048
#define DIN   256
#define DH    256
#define LDSW  260   // padded LDS row stride (260 % 64 == 4 -> conflict-free b64 A-reads)

// A-operand fetch: lane half splits K within a 4-wide chunk (ISA 16x4 f32 layout)
#define LDA(base, j) (*(const v2f*)(&(base)[lmod * LDSW + 4 * (j) + 2 * lhalf]))

// Scheduler pipeline spec for one 64-WMMA / 32-DS-read chain:
// 6 DS reads run ahead, then steady-state 2 WMMA : 1 DS read, 12-WMMA tail.
// mask 0x100 = DS read, mask 0x008 = MFMA/WMMA.
__device__ __forceinline__ void wmma_ds_pipeline_spec() {
  __builtin_amdgcn_sched_group_barrier(0x100, 6, 0);
#pragma unroll
  for (int q = 0; q < 26; ++q) {
    __builtin_amdgcn_sched_group_barrier(0x008, 2, 0);
    __builtin_amdgcn_sched_group_barrier(0x100, 1, 0);
  }
  __builtin_amdgcn_sched_group_barrier(0x008, 12, 0);
}

// ---------------------------------------------------------------------------
// Phase 1: xproj[b,t,:] = x[b,t,:] @ W_xh^T + b_h   (written into d_out)
// ---------------------------------------------------------------------------
__global__ __launch_bounds__(512) void rnn_xproj_kernel(
    const float* __restrict__ x, const float* __restrict__ Wxh,
    const float* __restrict__ bh, float* __restrict__ out,
    int num_mtiles, int grid_wgs) {
  __shared__ float sx[16 * LDSW];
  const int tid   = threadIdx.x;
  const int lane  = tid & 31;
  const int wave  = tid >> 5;
  const int n0    = wave * 16;
  const int lhalf = lane >> 4;
  const int lmod  = lane & 15;

  // B-operand layout: wb[j].x = Wxh[n0+lmod][4j + 2*lhalf], .y = next col
  v2f wb[64];
  const float* wrow = Wxh + (size_t)(n0 + lmod) * DIN + 2 * lhalf;
#pragma unroll
  for (int j = 0; j < 64; ++j) wb[j] = *(const v2f*)(wrow + 4 * j);

  const float bias = bh[n0 + lmod];

  for (int mt = blockIdx.x; mt < num_mtiles; mt += grid_wgs) {
    const size_t row0 = (size_t)mt * 16;
    {  // cooperative coalesced load of x tile [16][256] -> padded LDS
      const float* src = x + row0 * DIN;
      const int e = tid * 8;
      const int r = e >> 8, c = e & 255;
      v4f q0 = *(const v4f*)(src + r * DIN + c);
      v4f q1 = *(const v4f*)(src + r * DIN + c + 4);
      *(v4f*)(&sx[r * LDSW + c])     = q0;
      *(v4f*)(&sx[r * LDSW + c + 4]) = q1;
    }
    __syncthreads();

    v8f c8;
#pragma unroll
    for (int v = 0; v < 8; ++v) c8[v] = bias;

#pragma unroll
    for (int j = 0; j < 64; ++j) {
      v2f a = LDA(sx, j);
      c8 = __builtin_amdgcn_wmma_f32_16x16x4_f32(false, a, false, wb[j],
                                                 (short)0, c8, false, false);
    }
    wmma_ds_pipeline_spec();

    float* orow = out + (row0 + 8 * lhalf) * DH + n0 + lmod;
#pragma unroll
    for (int v = 0; v < 8; ++v) orow[(size_t)v * DH] = c8[v];

    __syncthreads();
  }
}

// ---------------------------------------------------------------------------
// Phase 2: sequential scan. 4 WGs (16 batch rows each, independent).
// Double-buffered LDS h tile (1 barrier/step), prefetched xproj C-init,
// scheduler-directed DS/WMMA pipeline on the serial critical path.
// ---------------------------------------------------------------------------
__global__ __launch_bounds__(512) void rnn_recur_kernel(
    const float* __restrict__ Whh, float* __restrict__ out) {
  __shared__ float sh[2][16 * LDSW];
  const int tid   = threadIdx.x;
  const int lane  = tid & 31;
  const int wave  = tid >> 5;
  const int n0    = wave * 16;
  const int lhalf = lane >> 4;
  const int lmod  = lane & 15;
  const int b0    = blockIdx.x * 16;

  v2f wb[64];
  const float* wrow = Whh + (size_t)(n0 + lmod) * DH + 2 * lhalf;
#pragma unroll
  for (int j = 0; j < 64; ++j) wb[j] = *(const v2f*)(wrow + 4 * j);

  for (int i = tid; i < 16 * LDSW; i += 512) sh[0][i] = 0.0f;  // h_0 = 0
  __syncthreads();

  float* obase = out + ((size_t)(b0 + 8 * lhalf) * SEQT) * DH + n0 + lmod;
  const size_t rstride = (size_t)SEQT * DH;  // +1 batch row

  v8f xp;  // prefetched xproj C-init for current t
#pragma unroll
  for (int v = 0; v < 8; ++v) xp[v] = obase[(size_t)v * rstride];  // t = 0

#pragma unroll 1
  for (int t = 0; t < SEQT; ++t) {
    const float* hb = sh[t & 1];        // read h_{t-1}
    float*       hw = sh[(t + 1) & 1];  // write h_t
    float*       op = obase + (size_t)t * DH;

    v8f c8 = xp;

    // prefetch next step's xproj during the WMMA chain (clamped, branch-free
    // so the loop body stays one scheduling region)
    {
      const int tn = (t < SEQT - 1) ? (t + 1) : t;
      const float* opn = obase + (size_t)tn * DH;
#pragma unroll
      for (int v = 0; v < 8; ++v) xp[v] = opn[(size_t)v * rstride];
    }

#pragma unroll
    for (int j = 0; j < 64; ++j) {
      v2f a = LDA(hb, j);
      c8 = __builtin_amdgcn_wmma_f32_16x16x4_f32(false, a, false, wb[j],
                                                 (short)0, c8, false, false);
    }
    wmma_ds_pipeline_spec();

#pragma unroll
    for (int v = 0; v < 8; ++v) c8[v] = fmaxf(c8[v], 0.0f);

#pragma unroll
    for (int v = 0; v < 8; ++v) {
      hw[(v + 8 * lhalf) * LDSW + n0 + lmod] = c8[v];  // h_t for next step
      op[(size_t)v * rstride] = c8[v];                  // output h_t
    }
    __syncthreads();  // h_t visible; also fences buffer reuse
  }
}

extern "C" void kernel_launch(void* const* d_in, const int* in_sizes, int n_in,
                              void* d_out, int out_size, void* d_ws, size_t ws_size,
                              hipStream_t stream) {
  const float* x   = (const float*)d_in[0];
  const float* Wxh = (const float*)d_in[1];
  const float* Whh = (const float*)d_in[2];
  const float* bh  = (const float*)d_in[3];
  float* out = (float*)d_out;

  const int num_mtiles = (BATCH * SEQT) / 16;  // 8192
  const int grid1 = 1024;
  rnn_xproj_kernel<<<grid1, 512, 0, stream>>>(x, Wxh, bh, out, num_mtiles, grid1);
  rnn_recur_kernel<<<BATCH / 16, 512, 0, stream>>>(Whh, out);
}